// FeatureSingleHeadAttention_61426622267528
// MI455X (gfx1250) — compile-verified
//
#include <hip/hip_runtime.h>
#include <math.h>

// Problem constants (from reference): B=1024, N=16 neighbors (+1 self = 17 blocks),
// T=50, H=128, TEMPERATURE=1.
#define B_SZ   1024
#define NBLK   17
#define T_SZ   50
#define H_SZ   128
#define ROWS   (NBLK * T_SZ)   // 850 KV rows per batch

typedef __attribute__((ext_vector_type(2))) float v2f;
typedef __attribute__((ext_vector_type(4))) float v4f;
typedef __attribute__((ext_vector_type(8))) float v8f;

// ---------------------------------------------------------------------------
// C[M,128] = A[M,128] * B[128,128]   (fp32 WMMA 16x16x4, one wave per 16x16 tile)
//   A row m element k read at A[m*lda + coff + k]   (coff = ts*H for kernel 1)
//   TRANSB=1 reads B as B^T (element (k,n) at Bm[n*128+k]) -> computes A*W^T
//   TRANSB is a template arg so the K-loop is branch-free straight-line code.
// ---------------------------------------------------------------------------
template <int TRANSB>
__global__ void gemm128_wmma(const float* __restrict__ A, int lda,
                             const int* __restrict__ tsp,
                             const float* __restrict__ Bm,
                             float* __restrict__ C) {
    const int lane = threadIdx.x;        // 0..31 (wave32)
    const int mt   = blockIdx.x;         // 16-row tile of M
    const int nt   = blockIdx.y;         // 16-col tile of N (128/16 = 8)
    const int coff = tsp ? tsp[0] * H_SZ : 0;
    const int g    = lane >> 4;          // lane group: 0 => K pair {0,1}, 1 => {2,3}
    const int hl   = lane & 15;

    // Hoisted base pointers: per-iteration address math is just +constant.
    const float* __restrict__ arow = A + (size_t)(mt * 16 + hl) * lda + coff + 2 * g;
    const float* __restrict__ bptr = TRANSB
        ? Bm + (size_t)(nt * 16 + hl) * H_SZ + 2 * g       // B^T row = ncol
        : Bm + (size_t)(2 * g) * H_SZ + nt * 16 + hl;      // B row = k

    v8f acc = {};
    #pragma unroll
    for (int k0 = 0; k0 < H_SZ; k0 += 4) {
        v2f a, b;
        // A 16x4 fragment: lanes0-15 hold K=k0,k0+1; lanes16-31 hold K=k0+2,k0+3
        a.x = arow[k0];
        a.y = arow[k0 + 1];
        if (TRANSB) {
            b.x = bptr[k0];
            b.y = bptr[k0 + 1];
        } else {
            b.x = bptr[(size_t)k0 * H_SZ];
            b.y = bptr[(size_t)(k0 + 1) * H_SZ];
        }
        // (neg_a, A, neg_b, B, c_mod, C, reuse_a, reuse_b)
        acc = __builtin_amdgcn_wmma_f32_16x16x4_f32(
            false, a, false, b, (short)0, acc, false, false);
    }
    // C/D layout: VGPR r -> lanes0-15: M=r, lanes16-31: M=r+8; N = lane&15
    #pragma unroll
    for (int r = 0; r < 8; ++r) {
        const int row = mt * 16 + r + g * 8;
        C[(size_t)row * H_SZ + nt * 16 + hl] = acc[r];
    }
}

// ---------------------------------------------------------------------------
// One workgroup per batch. Streams all 850 KV rows once (445 MB total across
// the grid -> HBM-bandwidth bound, ~19us at 23.3 TB/s). kv loads are
// non-temporal (read-once stream >> 192MB L2). Online softmax keeps running
// (m, l, acc[128]) per wave; raw scores written to the A output region; the
// 8 waves merge via LDS.
// ---------------------------------------------------------------------------
__global__ void attn_online(const float* __restrict__ node,    // [B,T,H]
                            const float* __restrict__ neigh,   // [B,16,T,H]
                            const unsigned char* __restrict__ mask, // [B,T] bool
                            const float* __restrict__ kq,      // [B,H] = Wk @ q
                            float* __restrict__ Araw,          // [B,850] raw scores out
                            float* __restrict__ ctx,           // [B,H] weighted kv
                            float* __restrict__ mout,          // [B] softmax max
                            float* __restrict__ lout) {        // [B] softmax denom
    const int b    = blockIdx.x;
    const int tid  = threadIdx.x;
    const int w    = tid >> 5;     // wave id 0..7
    const int lane = tid & 31;

    __shared__ float s_m[8];
    __shared__ float s_l[8];
    __shared__ float s_acc[8 * H_SZ];

    const float4 kq4 = ((const float4*)(kq + (size_t)b * H_SZ))[lane];

    float m    = -INFINITY;
    float lsum = 0.0f;
    v4f   acc  = {};

    for (int r = w; r < ROWS; r += 8) {
        const int n = r / T_SZ;
        const int t = r - n * T_SZ;
        const float* rowp = (n == 0)
            ? node  + ((size_t)b * T_SZ + t) * H_SZ
            : neigh + ((((size_t)b * (NBLK - 1)) + (n - 1)) * T_SZ + t) * H_SZ;

        // 32 lanes x 16B = one coalesced 512B row; TH=NT keeps L2 clean.
        const v4f x = __builtin_nontemporal_load((const v4f*)rowp + lane);
        float p = x.x * kq4.x + x.y * kq4.y + x.z * kq4.z + x.w * kq4.w;
        #pragma unroll
        for (int off = 16; off >= 1; off >>= 1)
            p += __shfl_xor(p, off, 32);
        // p now holds the full dot product S[b,n,t] in every lane

        const bool msk = mask[b * T_SZ + t] != 0;
        if (lane == 0)
            Araw[(size_t)b * ROWS + r] = msk ? -INFINITY : p;

        if (!msk) {
            const float mn   = fmaxf(m, p);
            const float corr = __expf(m - mn);   // m==-inf -> 0 (safe)
            const float e    = __expf(p - mn);
            lsum = lsum * corr + e;
            acc  = acc * corr + e * x;
            m = mn;
        }
    }

    if (lane == 0) { s_m[w] = m; s_l[w] = lsum; }
    ((v4f*)(s_acc + w * H_SZ))[lane] = acc;
    __syncthreads();

    if (w == 0) {
        float gm = -INFINITY;
        #pragma unroll
        for (int i = 0; i < 8; ++i) gm = fmaxf(gm, s_m[i]);
        float gl = 0.0f;
        v4f   ga = {};
        #pragma unroll
        for (int i = 0; i < 8; ++i) {
            if (s_m[i] > -INFINITY) {
                const float c = __expf(s_m[i] - gm);
                const v4f   a = ((const v4f*)(s_acc + i * H_SZ))[lane];
                gl += s_l[i] * c;
                ga  = ga + a * c;
            }
        }
        const float inv = 1.0f / gl;
        ((v4f*)(ctx + (size_t)b * H_SZ))[lane] = ga * inv;
        if (lane == 0) { mout[b] = gm; lout[b] = gl; }
    }
}

// A[idx] = exp(S - m[b]) / l[b]   (masked rows stored -inf -> exp gives 0)
__global__ void norm_A(float* __restrict__ A,
                       const float* __restrict__ m,
                       const float* __restrict__ l) {
    const int idx = blockIdx.x * blockDim.x + threadIdx.x;
    if (idx >= B_SZ * ROWS) return;
    const int b = idx / ROWS;
    A[idx] = __expf(A[idx] - m[b]) / l[b];
}

extern "C" void kernel_launch(void* const* d_in, const int* in_sizes, int n_in,
                              void* d_out, int out_size, void* d_ws, size_t ws_size,
                              hipStream_t stream) {
    const float*         node  = (const float*)d_in[0];
    const float*         neigh = (const float*)d_in[1];
    const unsigned char* msk   = (const unsigned char*)d_in[2];
    const float*         Wq    = (const float*)d_in[3];
    const float*         Wk    = (const float*)d_in[4];
    const float*         Wv    = (const float*)d_in[5];
    const int*           tsp   = (const int*)d_in[6];   // device scalar

    float* out  = (float*)d_out;             // [B,H] attention output
    float* outA = out + B_SZ * H_SZ;         // [B,17,50] attention weights

    float* ws   = (float*)d_ws;
    float* q    = ws;                        // [B,H]
    float* kq   = q    + B_SZ * H_SZ;        // [B,H]
    float* ctx  = kq   + B_SZ * H_SZ;        // [B,H]
    float* mbuf = ctx  + B_SZ * H_SZ;        // [B]
    float* lbuf = mbuf + B_SZ;               // [B]

    dim3 gg(B_SZ / 16, H_SZ / 16);           // 64 x 8 tiles, 1 wave each

    // 1) q = node[:, ts, :] @ Wq            (fp32 WMMA)
    gemm128_wmma<0><<<gg, 32, 0, stream>>>(node, T_SZ * H_SZ, tsp, Wq, q);
    // 2) kq = q @ Wk^T                      (fp32 WMMA)
    gemm128_wmma<1><<<gg, 32, 0, stream>>>(q, H_SZ, nullptr, Wk, kq);
    // 3) single streaming pass: scores + online softmax + ctx accumulation
    attn_online<<<B_SZ, 256, 0, stream>>>(node, neigh, msk, kq, outA, ctx, mbuf, lbuf);
    // 4) normalize attention weights in-place
    norm_A<<<(B_SZ * ROWS + 255) / 256, 256, 0, stream>>>(outA, mbuf, lbuf);
    // 5) out = ctx @ Wv                     (fp32 WMMA)
    gemm128_wmma<0><<<gg, 32, 0, stream>>>(ctx, H_SZ, nullptr, Wv, out);
}